// GraphConvLayer_71889162600963
// MI455X (gfx1250) — compile-verified
//
#include <hip/hip_runtime.h>

typedef __attribute__((ext_vector_type(2))) float v2f;
typedef __attribute__((ext_vector_type(8))) float v8f;

#define N_NODES 100000
#define N_EDGES 1200000
#define D_IN 64
#define D_OUT 64

// ---------------------------------------------------------------------------
// Stage 1: support = x @ W using V_WMMA_F32_16X16X4_F32 (full fp32 precision).
// One wave32 computes a 16-row x 64-col output stripe:
//   4 accumulators (16x16 f32 each, 8 VGPRs), K stepped by 4 -> 64 WMMAs/wave.
// N_NODES = 6250 * 16 exactly, so no row remainder handling needed.
// ---------------------------------------------------------------------------
__global__ __launch_bounds__(256)
void gcn_gemm_wmma(const float* __restrict__ x,
                   const float* __restrict__ w,
                   float* __restrict__ support) {
    const int wave = (int)((blockIdx.x * blockDim.x + threadIdx.x) >> 5);
    const int lane = (int)(threadIdx.x & 31);
    // Wave-uniform exit: EXEC remains all-ones inside active waves (WMMA req).
    if (wave >= N_NODES / 16) return;

    const int rb = wave * 16;      // row-tile base
    const int lh = lane & 15;      // 0..15
    const int hi = lane >> 4;      // 0 (K=0,1) or 1 (K=2,3) for A/B tiles

    v8f acc0 = {}, acc1 = {}, acc2 = {}, acc3 = {};

    const float* xrow = x + (size_t)(rb + lh) * D_IN;

    #pragma unroll
    for (int k = 0; k < D_IN; k += 4) {
        // A tile 16x4: lane lh holds row rb+lh; vgpr0=K=k+2*hi, vgpr1=K+1
        v2f a;
        a.x = xrow[k + 2 * hi + 0];
        a.y = xrow[k + 2 * hi + 1];

        // B tiles 4x16 for each of the 4 column tiles:
        // lane lh holds column cb+lh; vgpr0=K=k+2*hi, vgpr1=K+1
        v2f b0, b1, b2, b3;
        const float* wr0 = w + (size_t)(k + 2 * hi + 0) * D_OUT + lh;
        const float* wr1 = w + (size_t)(k + 2 * hi + 1) * D_OUT + lh;
        b0.x = wr0[0];  b0.y = wr1[0];
        b1.x = wr0[16]; b1.y = wr1[16];
        b2.x = wr0[32]; b2.y = wr1[32];
        b3.x = wr0[48]; b3.y = wr1[48];

        acc0 = __builtin_amdgcn_wmma_f32_16x16x4_f32(false, a, false, b0,
                                                     (short)0, acc0, false, false);
        acc1 = __builtin_amdgcn_wmma_f32_16x16x4_f32(false, a, false, b1,
                                                     (short)0, acc1, false, false);
        acc2 = __builtin_amdgcn_wmma_f32_16x16x4_f32(false, a, false, b2,
                                                     (short)0, acc2, false, false);
        acc3 = __builtin_amdgcn_wmma_f32_16x16x4_f32(false, a, false, b3,
                                                     (short)0, acc3, false, false);
    }

    // C/D 16x16 f32 layout: vgpr j, lanes 0-15 -> M=j, lanes 16-31 -> M=j+8,
    // N = lane%16.  Store the 16x64 stripe.
    #pragma unroll
    for (int j = 0; j < 8; ++j) {
        float* orow = support + (size_t)(rb + hi * 8 + j) * D_OUT + lh;
        orow[0]  = acc0[j];
        orow[16] = acc1[j];
        orow[32] = acc2[j];
        orow[48] = acc3[j];
    }
}

// ---------------------------------------------------------------------------
// Stage 2: out[i][f] = bias[f]   (accumulation base for the scatter pass)
// ---------------------------------------------------------------------------
__global__ __launch_bounds__(256)
void gcn_bias_init(const float* __restrict__ bias, float* __restrict__ out) {
    const long long i = (long long)blockIdx.x * blockDim.x + threadIdx.x;
    if (i < (long long)N_NODES * D_OUT) {
        out[i] = bias[i & (D_OUT - 1)];
    }
}

// ---------------------------------------------------------------------------
// Stage 3: out[row[e]] += val[e] * support[col[e]]
// One thread per (edge, feature).  Each wave32 shares a single edge
// (t>>6 is wave-uniform), so row/col/val loads broadcast; the 32 feature
// lanes are coalesced.  Float atomics resolve in L2 (support/out L2-resident).
// ---------------------------------------------------------------------------
__global__ __launch_bounds__(256)
void gcn_edge_scatter(const float* __restrict__ support,
                      const float* __restrict__ edge_val,
                      const int* __restrict__ edge_row,
                      const int* __restrict__ edge_col,
                      float* __restrict__ out) {
    const long long t = (long long)blockIdx.x * blockDim.x + threadIdx.x;
    if (t >= (long long)N_EDGES * D_OUT) return;
    const int e = (int)(t >> 6);
    const int f = (int)(t & (D_OUT - 1));

    const int r = edge_row[e];
    const int c = edge_col[e];
    const float v = edge_val[e];

    const float m = v * support[(size_t)c * D_OUT + f];
    atomicAdd(&out[(size_t)r * D_OUT + f], m);
}

// ---------------------------------------------------------------------------
extern "C" void kernel_launch(void* const* d_in, const int* in_sizes, int n_in,
                              void* d_out, int out_size, void* d_ws, size_t ws_size,
                              hipStream_t stream) {
    const float* x        = (const float*)d_in[0];
    const float* weight   = (const float*)d_in[1];
    const float* bias     = (const float*)d_in[2];
    const float* edge_val = (const float*)d_in[3];
    const int*   edge_row = (const int*)d_in[4];
    const int*   edge_col = (const int*)d_in[5];
    float*       out      = (float*)d_out;
    float*       support  = (float*)d_ws;   // N_NODES * D_OUT floats = 25.6 MB

    // Stage 1: dense projection via WMMA.  6250 waves, 8 waves per block.
    {
        const int waves  = N_NODES / 16;               // 6250
        const int blocks = (waves + 7) / 8;            // 782
        gcn_gemm_wmma<<<blocks, 256, 0, stream>>>(x, weight, support);
    }

    // Stage 2: initialize output with bias.
    {
        const long long n = (long long)N_NODES * D_OUT;
        const int blocks = (int)((n + 255) / 256);
        gcn_bias_init<<<blocks, 256, 0, stream>>>(bias, out);
    }

    // Stage 3: edge gather/scale/scatter with float atomics.
    {
        const long long n = (long long)N_EDGES * D_OUT;
        const int blocks = (int)((n + 255) / 256);
        gcn_edge_scatter<<<blocks, 256, 0, stream>>>(support, edge_val,
                                                     edge_row, edge_col, out);
    }
}